// GATLayer_13640816132543
// MI455X (gfx1250) — compile-verified
//
#include <hip/hip_runtime.h>
#include <hip/hip_bf16.h>

typedef __attribute__((ext_vector_type(2))) float v2f;
typedef __attribute__((ext_vector_type(8))) float v8f;

#define LRELU_A 0.2f
#define Bn   4
#define NHn  4
#define Nn   2048
#define INd  128
#define OUTd 32
#define LDA  2052   // padded LDS row stride (floats): 2052*4 % 256 = 16 -> conflict-free column reads

// ---------------------------------------------------------------------------
// Kernel 1: Ht[b][h][n][f] = sum_i H[b][n][i] * W[h][i][f]
// One wave per 16-row x 32-col output tile, V_WMMA_F32_16X16X4_F32, K-loop 128/4.
// ---------------------------------------------------------------------------
__global__ __launch_bounds__(128) void gat_proj_kernel(
    const float* __restrict__ H, const float* __restrict__ W,
    float* __restrict__ Ht)
{
    const int lane = threadIdx.x & 31;
    const int wave = threadIdx.x >> 5;
    const int gw   = blockIdx.x * 4 + wave;      // 0 .. B*NH*(N/16)-1 = 2047
    const int mtile = gw & (Nn / 16 - 1);        // 128 tiles per (b,h)
    const int bh    = gw >> 7;                   // 0..15
    const int b = bh >> 2, h = bh & 3;

    const float* __restrict__ Hb = H + ((size_t)b * Nn + (size_t)mtile * 16) * INd;
    const float* __restrict__ Wb = W + (size_t)h * INd * OUTd;

    const int m  = lane & 15;            // A row
    const int kb = (lane >> 4) * 2;      // A/B K sub-offset (K=0/1 vs K=2/3)
    const int n0 = lane & 15;            // B/C column

    v8f acc0 = {}; v8f acc1 = {};
    for (int k0 = 0; k0 < INd; k0 += 4) {
        v2f af; af.x = Hb[m * INd + k0 + kb];
                af.y = Hb[m * INd + k0 + kb + 1];
        v2f b0; b0.x = Wb[(k0 + kb) * OUTd + n0];
                b0.y = Wb[(k0 + kb + 1) * OUTd + n0];
        v2f b1; b1.x = Wb[(k0 + kb) * OUTd + n0 + 16];
                b1.y = Wb[(k0 + kb + 1) * OUTd + n0 + 16];
        acc0 = __builtin_amdgcn_wmma_f32_16x16x4_f32(false, af, false, b0, (short)0, acc0, false, false);
        acc1 = __builtin_amdgcn_wmma_f32_16x16x4_f32(false, af, false, b1, (short)0, acc1, false, false);
    }

    float* __restrict__ HtB = Ht + ((size_t)bh * Nn + (size_t)mtile * 16) * OUTd;
    const int rbase = (lane < 16) ? 0 : 8;
    #pragma unroll
    for (int r = 0; r < 8; ++r) {
        HtB[(rbase + r) * OUTd + n0]      = acc0[r];
        HtB[(rbase + r) * OUTd + n0 + 16] = acc1[r];
    }
}

// ---------------------------------------------------------------------------
// Kernel 2: src[bh][n] = Ht[bh][n][:] . a_src[h] ; dst likewise
// ---------------------------------------------------------------------------
__global__ __launch_bounds__(256) void gat_coef_kernel(
    const float* __restrict__ Ht, const float* __restrict__ a,
    float* __restrict__ srcv, float* __restrict__ dstv)
{
    const int tid = blockIdx.x * 256 + threadIdx.x;   // 0 .. B*NH*N-1
    const int h = (tid >> 11) & 3;
    const float* __restrict__ row = Ht + (size_t)tid * OUTd;
    const float* __restrict__ as  = a + h * 2 * OUTd;
    const float* __restrict__ ad  = as + OUTd;
    float s = 0.f, d = 0.f;
    #pragma unroll
    for (int f = 0; f < OUTd; ++f) {
        float v = row[f];
        s += v * as[f];
        d += v * ad[f];
    }
    srcv[tid] = s;
    dstv[tid] = d;
}

// ---------------------------------------------------------------------------
// Kernel 3 (fused): for one (b,h) and a 16-row tile:
//   e = leakyrelu(src_i + dst_j) masked by adj+I  -> LDS (padded)
//   row softmax (16 threads/row, LDS reductions)
//   alpha -> d_out with NON-TEMPORAL stores (268 MB stream; keep adj/Ht in L2)
//   split-K WMMA across 8 waves: 16x2048 @ 2048x32, partials reduced in LDS
//   ELU-ish activation, transposed NT store to H_new
// Dynamic LDS: 16*2052 alpha + 8*512 partials + 256 reduce + 16 inv = 148,800 B
// (fits the 320KB CDNA5 WGP LDS; impossible on a 64KB-LDS part)
// ---------------------------------------------------------------------------
__global__ __launch_bounds__(256) void gat_softmax_agg_kernel(
    const float* __restrict__ Ht, const float* __restrict__ srcv,
    const float* __restrict__ dstv, const float* __restrict__ adj,
    float* __restrict__ alphaOut, float* __restrict__ Hnew)
{
    extern __shared__ float smem[];
    float* sAlpha = smem;                  // 16 * LDA
    float* sPart  = smem + 16 * LDA;       // 8 * 512
    float* sRed   = sPart + 8 * 512;       // 256
    float* sInv   = sRed + 256;            // 16

    const int tid   = threadIdx.x;
    const int itile = blockIdx.x & (Nn / 16 - 1);
    const int bh    = blockIdx.x >> 7;     // 0..15
    const int b = bh >> 2, h = bh & 3;

    const int r = tid >> 4;                // row in tile, 0..15
    const int c = tid & 15;                // 16 threads per row

    const int ig = itile * 16 + r;         // global row index
    const float sv = srcv[(size_t)bh * Nn + ig];
    const float* __restrict__ dv   = dstv + (size_t)bh * Nn;
    const float* __restrict__ arow = adj + (size_t)ig * Nn;

    const float NEG = -__builtin_inff();

    // Pass 1: e = masked leakyrelu, track per-thread max
    float mx = NEG;
    for (int jj = 0; jj < Nn / 16; ++jj) {
        int j = c + jj * 16;
        float e = sv + dv[j];
        e = (e >= 0.f) ? e : LRELU_A * e;
        bool conn = (arow[j] != 0.f) || (ig == j);
        e = conn ? e : NEG;
        sAlpha[r * LDA + j] = e;
        mx = fmaxf(mx, e);
    }
    sRed[r * 16 + c] = mx;
    __syncthreads();
    float rowmax = NEG;
    #pragma unroll
    for (int t = 0; t < 16; ++t) rowmax = fmaxf(rowmax, sRed[r * 16 + t]);
    __syncthreads();

    // Pass 2: exp(e - max), per-thread sum
    float sum = 0.f;
    for (int jj = 0; jj < Nn / 16; ++jj) {
        int j = c + jj * 16;
        float p = __expf(sAlpha[r * LDA + j] - rowmax);
        sAlpha[r * LDA + j] = p;
        sum += p;
    }
    sRed[r * 16 + c] = sum;
    __syncthreads();
    float tot = 0.f;
    #pragma unroll
    for (int t = 0; t < 16; ++t) tot += sRed[r * 16 + t];
    if (c == 0) sInv[r] = 1.0f / tot;
    __syncthreads();

    // Pass 3: normalize; alpha -> global (row-major coalesced, NON-TEMPORAL:
    // alpha is write-once/never re-read from HBM, don't evict adj/Ht from L2)
    float* __restrict__ aOut = alphaOut + ((size_t)bh * Nn + (size_t)itile * 16) * Nn;
    for (int idx = tid; idx < 16 * Nn; idx += 256) {
        int row = idx >> 11;               // /Nn
        int j   = idx & (Nn - 1);
        float v = sAlpha[row * LDA + j] * sInv[row];
        sAlpha[row * LDA + j] = v;
        __builtin_nontemporal_store(v, &aOut[(size_t)row * Nn + j]);
    }
    __syncthreads();

    // Phase B: split-K WMMA. Wave w covers K in [w*256, (w+1)*256)
    const int lane = tid & 31;
    const int wv   = tid >> 5;             // 0..7
    const int m    = lane & 15;
    const int kb   = (lane >> 4) * 2;
    const int n0   = lane & 15;
    const float* __restrict__ HtB = Ht + (size_t)bh * Nn * OUTd;

    v8f acc0 = {}; v8f acc1 = {};
    const int k0base = wv * (Nn / 8);
    for (int ks = 0; ks < (Nn / 8) / 4; ++ks) {
        int k0 = k0base + ks * 4;
        v2f af; af.x = sAlpha[m * LDA + k0 + kb];
                af.y = sAlpha[m * LDA + k0 + kb + 1];
        v2f b0; b0.x = HtB[(k0 + kb) * OUTd + n0];
                b0.y = HtB[(k0 + kb + 1) * OUTd + n0];
        v2f b1; b1.x = HtB[(k0 + kb) * OUTd + n0 + 16];
                b1.y = HtB[(k0 + kb + 1) * OUTd + n0 + 16];
        acc0 = __builtin_amdgcn_wmma_f32_16x16x4_f32(false, af, false, b0, (short)0, acc0, false, false);
        acc1 = __builtin_amdgcn_wmma_f32_16x16x4_f32(false, af, false, b1, (short)0, acc1, false, false);
    }

    const int rbase = (lane < 16) ? 0 : 8;
    #pragma unroll
    for (int q = 0; q < 8; ++q) {
        sPart[wv * 512 + (rbase + q) * OUTd + n0]      = acc0[q];
        sPart[wv * 512 + (rbase + q) * OUTd + n0 + 16] = acc1[q];
    }
    __syncthreads();

    // Reduce 8 wave-partials, ELU-ish activation, transposed NT store
    for (int e = tid; e < 512; e += 256) {
        float s = 0.f;
        #pragma unroll
        for (int w = 0; w < 8; ++w) s += sPart[w * 512 + e];
        s = (s > 0.f) ? s : expm1f(s);
        int row = e >> 5, f = e & 31;
        __builtin_nontemporal_store(
            s, &Hnew[((size_t)b * Nn + (itile * 16 + row)) * (NHn * OUTd) + h * OUTd + f]);
    }
}

// ---------------------------------------------------------------------------
extern "C" void kernel_launch(void* const* d_in, const int* in_sizes, int n_in,
                              void* d_out, int out_size, void* d_ws, size_t ws_size,
                              hipStream_t stream) {
    (void)in_sizes; (void)n_in; (void)out_size; (void)ws_size;
    const float* H   = (const float*)d_in[0];   // (4,2048,128)
    const float* adj = (const float*)d_in[1];   // (2048,2048)
    const float* W   = (const float*)d_in[2];   // (4,128,32)
    const float* a   = (const float*)d_in[3];   // (4,64)

    float* out   = (float*)d_out;
    float* Hnew  = out;                                   // B*N*(NH*OUTd) = 1,048,576
    float* alpha = out + (size_t)Bn * Nn * NHn * OUTd;    // B*NH*N*N

    float* Ht   = (float*)d_ws;                           // B*NH*N*OUTd = 1,048,576 floats
    float* srcv = Ht + (size_t)Bn * NHn * Nn * OUTd;      // 32768
    float* dstv = srcv + (size_t)Bn * NHn * Nn;           // 32768

    gat_proj_kernel<<<512, 128, 0, stream>>>(H, W, Ht);
    gat_coef_kernel<<<(Bn * NHn * Nn) / 256, 256, 0, stream>>>(Ht, a, srcv, dstv);

    const size_t shmem = (size_t)(16 * LDA + 8 * 512 + 256 + 16) * sizeof(float);
    gat_softmax_agg_kernel<<<Bn * NHn * (Nn / 16), 256, shmem, stream>>>(
        Ht, srcv, dstv, adj, alpha, Hnew);
}